// Set2Set_GRU_20667382628743
// MI455X (gfx1250) — compile-verified
//
#include <hip/hip_runtime.h>
#include <math.h>

typedef __attribute__((ext_vector_type(2))) float v2f;
typedef __attribute__((ext_vector_type(8))) float v8f;

// ---------------------------------------------------------------------------
// C[M,N] = A[M,K] @ W[N,K]^T + bias[N]  (all f32, row-major)
// One wave computes one 16x16 tile with v_wmma_f32_16x16x4_f32 (exact fp32).
// A-frag layout (16x4 f32): lane -> m = lane%16, holds K = {2h, 2h+1}, h=lane/16
// B-frag layout (4x16 f32): lane -> n = lane%16, holds K = {2h, 2h+1}
// C/D (16x16 f32, 8 VGPRs): vgpr v, lane -> (m = v + 8h, n = lane%16)
// ---------------------------------------------------------------------------
__global__ void gemm_wmma_bias(const float* __restrict__ A,
                               const float* __restrict__ W,
                               const float* __restrict__ bias,
                               float* __restrict__ C,
                               int M, int N, int K)
{
    const int lane = threadIdx.x & 31;
    const int wave = threadIdx.x >> 5;
    const int wavesPerBlock = blockDim.x >> 5;
    const int tilesN = N >> 4;
    const int tilesM = M >> 4;
    const int tile = blockIdx.x * wavesPerBlock + wave;
    if (tile >= tilesM * tilesN) return;
    const int m0 = (tile / tilesN) << 4;
    const int n0 = (tile % tilesN) << 4;
    const int half = lane >> 4;
    const int l    = lane & 15;

    const float* arow = A + (size_t)(m0 + l) * K + 2 * half;
    const float* wrow = W + (size_t)(n0 + l) * K + 2 * half;

    v8f acc = {};
    for (int k = 0; k < K; k += 4) {
        v2f a = *(const v2f*)(arow + k);
        v2f b = *(const v2f*)(wrow + k);
        acc = __builtin_amdgcn_wmma_f32_16x16x4_f32(false, a, false, b,
                                                    (short)0, acc, false, false);
    }
    const int n = n0 + l;
    const float bn = bias[n];
#pragma unroll
    for (int v = 0; v < 8; ++v) {
        const int m = m0 + v + 8 * half;
        C[(size_t)m * N + n] = acc[v] + bn;
    }
}

// ---------------------------------------------------------------------------
// GRU gate combine: h' = (1-z)*n + z*h ; also writes q (=h') into q_star[:, :D]
// gate order (PyTorch): r, z, n
// ---------------------------------------------------------------------------
__global__ void gru_update(const float* __restrict__ gi,
                           const float* __restrict__ gh,
                           float* __restrict__ h,
                           float* __restrict__ q_star,
                           int B, int D)
{
    const int idx = blockIdx.x * blockDim.x + threadIdx.x;
    if (idx >= B * D) return;
    const int b = idx / D;
    const int c = idx - b * D;
    const float* gib = gi + (size_t)b * 3 * D;
    const float* ghb = gh + (size_t)b * 3 * D;

    const float r = 1.0f / (1.0f + __expf(-(gib[c] + ghb[c])));
    const float z = 1.0f / (1.0f + __expf(-(gib[D + c] + ghb[D + c])));
    const float n = tanhf(gib[2 * D + c] + r * ghb[2 * D + c]);
    const float hn = (1.0f - z) * n + z * h[idx];
    h[idx] = hn;
    q_star[(size_t)b * 2 * D + c] = hn;
}

// ---------------------------------------------------------------------------
// seg[b] = first node index with batch[i] >= b (batch is sorted); seg[B] = N
// ---------------------------------------------------------------------------
__global__ void segbounds(const int* __restrict__ batch, int* __restrict__ seg,
                          int N, int B)
{
    const int b = blockIdx.x * blockDim.x + threadIdx.x;
    if (b > B) return;
    if (b == B) { seg[B] = N; return; }
    int lo = 0, hi = N;
    while (lo < hi) {
        const int mid = (lo + hi) >> 1;
        if (batch[mid] < b) lo = mid + 1; else hi = mid;
    }
    seg[b] = lo;
}

__global__ void zero2(float* __restrict__ a, int na, float* __restrict__ b, int nb)
{
    const int i = blockIdx.x * blockDim.x + threadIdx.x;
    if (i < na) a[i] = 0.0f;
    if (i < nb) b[i] = 0.0f;
}

// ---------------------------------------------------------------------------
// Segment-softmax attention for one graph per block (256 thr = 8 waves).
// D must be 128: each lane owns a float4 slice of channels.
//   pass1: e[i] = dot(x_i, q_b), block max m
//   pass2: acc += exp(e-m)*x_i ; s += exp(e-m) ;  r = acc / (s + 1e-16)
// ---------------------------------------------------------------------------
__global__ void attention(const float* __restrict__ x,
                          const int* __restrict__ seg,
                          const float* __restrict__ h,
                          float* __restrict__ e,
                          float* __restrict__ q_star,
                          int D)
{
    const int b     = blockIdx.x;
    const int start = seg[b];
    const int end   = seg[b + 1];
    const int tid   = threadIdx.x;
    const int lane  = tid & 31;
    const int wave  = tid >> 5;
    const int NW    = blockDim.x >> 5;

    __shared__ float q[128];
    __shared__ float wmax[8];
    __shared__ float wsum[8];
    __shared__ float waveacc[8][128];

    if (tid < 128) q[tid] = h[(size_t)b * D + tid];
    __syncthreads();

    const float4 qv = ((const float4*)q)[lane];

    // ---- pass 1: dot products + segment max ----
    float lmax = -INFINITY;
    for (int i = start + wave; i < end; i += NW) {
        const float4 xv = ((const float4*)(x + (size_t)i * D))[lane];
        float d = xv.x * qv.x + xv.y * qv.y + xv.z * qv.z + xv.w * qv.w;
#pragma unroll
        for (int off = 16; off > 0; off >>= 1)
            d += __shfl_xor(d, off, 32);
        if (lane == 0) e[i] = d;
        lmax = fmaxf(lmax, d);
    }
    if (lane == 0) wmax[wave] = lmax;
    __syncthreads();
    if (tid == 0) {
        float m = wmax[0];
        for (int w = 1; w < NW; ++w) m = fmaxf(m, wmax[w]);
        wmax[0] = m;
    }
    __syncthreads();
    const float m = wmax[0];

    // ---- pass 2: softmax numerator vector and denominator ----
    float4 acc = make_float4(0.f, 0.f, 0.f, 0.f);
    float  sl  = 0.f;
    for (int i = start + wave; i < end; i += NW) {
        const float ex = __expf(e[i] - m);
        const float4 xv = ((const float4*)(x + (size_t)i * D))[lane];
        acc.x += ex * xv.x; acc.y += ex * xv.y;
        acc.z += ex * xv.z; acc.w += ex * xv.w;
        sl += ex;
    }
    ((float4*)waveacc[wave])[lane] = acc;
    if (lane == 0) wsum[wave] = sl;
    __syncthreads();

    if (tid < 128) {
        float r = 0.f;
        for (int w = 0; w < NW; ++w) r += waveacc[w][tid];
        float s = 0.f;
        for (int w = 0; w < NW; ++w) s += wsum[w];
        q_star[(size_t)b * 2 * D + D + tid] = r / (s + 1e-16f);
    }
}

// ---------------------------------------------------------------------------
extern "C" void kernel_launch(void* const* d_in, const int* in_sizes, int n_in,
                              void* d_out, int out_size, void* d_ws, size_t ws_size,
                              hipStream_t stream)
{
    (void)n_in; (void)ws_size;
    const float* x     = (const float*)d_in[0];
    const int*   batch = (const int*)  d_in[1];
    // d_in[2] = batch_size (unused; derived from out_size)
    const float* W_ih  = (const float*)d_in[3];
    const float* W_hh  = (const float*)d_in[4];
    const float* b_ih  = (const float*)d_in[5];
    const float* b_hh  = (const float*)d_in[6];
    float* q_star = (float*)d_out;

    const int N  = in_sizes[1];      // nodes
    const int D3 = in_sizes[5];      // 3*D
    const int D  = D3 / 3;           // 128
    const int B  = out_size / (2 * D);

    float* h  = (float*)d_ws;                 // [B, D]
    float* gi = h  + (size_t)B * D;           // [B, 3D]
    float* gh = gi + (size_t)B * D3;          // [B, 3D]
    float* e  = gh + (size_t)B * D3;          // [N]
    int*  seg = (int*)(e + N);                // [B+1]

    // zero initial h and q_star (deterministic across replays)
    const int nz = B * 2 * D;
    zero2<<<(nz + 255) / 256, 256, 0, stream>>>(h, B * D, q_star, nz);
    segbounds<<<(B + 1 + 255) / 256, 256, 0, stream>>>(batch, seg, N, B);

    const int tiles = (B / 16) * (D3 / 16);
    const int gemmBlocks = (tiles + 3) / 4;   // 4 waves per 128-thread block

    for (int step = 0; step < 3; ++step) {
        // gi = q_star @ W_ih^T + b_ih   (K = 2D)
        gemm_wmma_bias<<<gemmBlocks, 128, 0, stream>>>(q_star, W_ih, b_ih, gi, B, D3, 2 * D);
        // gh = h @ W_hh^T + b_hh        (K = D)
        gemm_wmma_bias<<<gemmBlocks, 128, 0, stream>>>(h, W_hh, b_hh, gh, B, D3, D);
        // h = GRU(q_star, h); q_star[:, :D] = h
        gru_update<<<(B * D + 255) / 256, 256, 0, stream>>>(gi, gh, h, q_star, B, D);
        // q_star[:, D:] = segment-softmax-weighted sum of x
        attention<<<B, 256, 0, stream>>>(x, seg, h, e, q_star, D);
    }
}